// L2LossWithRebalancing_65335042506927
// MI455X (gfx1250) — compile-verified
//
#include <hip/hip_runtime.h>

typedef __attribute__((ext_vector_type(2))) float v2f;
typedef __attribute__((ext_vector_type(8))) float v8f;

#define HW_SZ     65536        // 256*256
#define K_CENT    313
#define NTILES    20           // ceil(313/16)
#define NPIX      1048576      // 16*256*256
#define NGROUPS   (NPIX / 16)

__global__ void zero_out_kernel(float* out) {
    if (threadIdx.x == 0 && blockIdx.x == 0) out[0] = 0.0f;
}

__global__ __launch_bounds__(256)
void rebalanced_l2_kernel(const float* __restrict__ pred,
                          const float* __restrict__ targ,
                          const float* __restrict__ centers,
                          const float* __restrict__ weights,
                          float* __restrict__ out)
{
    __shared__ float s_w[320];
    for (int i = threadIdx.x; i < 320; i += 256)
        s_w[i] = (i < K_CENT) ? weights[i] : 0.0f;
    __syncthreads();

    const int  lane = threadIdx.x & 31;
    const int  m16  = lane & 15;
    const bool lo   = (lane < 16);

    // ---- Precompute A tiles (centers), kept in VGPRs across the pixel loop.
    // 32-bit A 16x4 layout: lanes 0-15 = K0,K1 ; lanes 16-31 = K2,K3.
    // K0=cx, K1=cy, K2=-0.5*|c|^2 (bias), K3=0.
    v2f Atile[NTILES];
#pragma unroll
    for (int t = 0; t < NTILES; ++t) {
        int  idx   = t * 16 + m16;
        bool valid = (idx < K_CENT);
        int  cidx  = valid ? idx : (K_CENT - 1);
        float cx   = centers[2 * cidx + 0];
        float cy   = centers[2 * cidx + 1];
        float bias = valid ? (-0.5f * (cx * cx + cy * cy)) : -1.0e30f;
        Atile[t].x = lo ? (valid ? cx : 0.0f) : bias;
        Atile[t].y = lo ? (valid ? cy : 0.0f) : 0.0f;
    }

    const int wavesPerBlock = blockDim.x >> 5;
    const int wave   = blockIdx.x * wavesPerBlock + (threadIdx.x >> 5);
    const int nwaves = gridDim.x * wavesPerBlock;

    float acc = 0.0f;

    for (int g = wave; g < NGROUPS; g += nwaves) {
        int p     = g * 16 + m16;          // flat pixel index (lanes 16-31 mirror 0-15)
        int bimg  = p >> 16;               // / HW_SZ
        int hw    = p & (HW_SZ - 1);
        int base  = bimg * 2 * HW_SZ + hw; // channel 0 of [B,2,H,W]

        float ta = targ[base];
        float tb = targ[base + HW_SZ];
        float pa = pred[base];
        float pb = pred[base + HW_SZ];

        // ---- B operand (pixels): 32-bit B 4x16 layout mirrors A:
        // VGPR0: lanes 0-15 = K0 row (128*a), lanes 16-31 = K2 row (1.0)
        // VGPR1: lanes 0-15 = K1 row (128*b), lanes 16-31 = K3 row (0.0)
        v2f Bop;
        Bop.x = lo ? (ta * 128.0f) : 1.0f;
        Bop.y = lo ? (tb * 128.0f) : 0.0f;

        float best = -3.0e38f;
        int   bidx = 0;

#pragma unroll
        for (int t = 0; t < NTILES; ++t) {
            v8f c = {};
            // D[m,n] = x_n . c_m - 0.5*|c_m|^2  (score; argmax == nearest center)
            v8f d = __builtin_amdgcn_wmma_f32_16x16x4_f32(
                        false, Atile[t], false, Bop, (short)0, c, false, false);
            int mbase = t * 16 + (lo ? 0 : 8);
#pragma unroll
            for (int r = 0; r < 8; ++r) {
                float s  = d[r];
                int   mi = mbase + r;
                bool better = (s > best);
                best = better ? s  : best;
                bidx = better ? mi : bidx;
            }
        }

        // Combine lane l with lane l^16 (rows 0-7 vs 8-15 of each tile).
        float obest = __shfl_xor(best, 16, 32);
        int   oidx  = __shfl_xor(bidx, 16, 32);
        bool better = (obest > best) || (obest == best && oidx < bidx);
        best = better ? obest : best;
        bidx = better ? oidx  : bidx;

        float w  = s_w[bidx];
        float da = pa - ta, db = pb - tb;
        float l2 = da * da + db * db;
        acc += lo ? (l2 * w) : 0.0f;       // lanes 16-31 are mirrors; contribute 0
    }

    acc *= (1.0f / (float)NPIX);

    // Wave reduction, one atomic per wave.
#pragma unroll
    for (int off = 1; off < 32; off <<= 1)
        acc += __shfl_xor(acc, off, 32);
    if (lane == 0) atomicAdd(out, acc);
}

extern "C" void kernel_launch(void* const* d_in, const int* in_sizes, int n_in,
                              void* d_out, int out_size, void* d_ws, size_t ws_size,
                              hipStream_t stream) {
    const float* pred    = (const float*)d_in[0]; // [16,2,256,256]
    const float* targ    = (const float*)d_in[1]; // [16,2,256,256]
    const float* centers = (const float*)d_in[2]; // [313,2]
    const float* weights = (const float*)d_in[3]; // [313]
    float* out = (float*)d_out;

    zero_out_kernel<<<1, 32, 0, stream>>>(out);
    rebalanced_l2_kernel<<<512, 256, 0, stream>>>(pred, targ, centers, weights, out);
}